// TransformedInput_46454366273939
// MI455X (gfx1250) — compile-verified
//
#include <hip/hip_runtime.h>
#include <hip/hip_bf16.h>
#include <stdint.h>

// Problem constants (x: [3,64,64] fp32)
#define CCH 3
#define HH  64
#define WW  64
#define NN  (CCH * HH * WW)      /* 12288 */
#define HWW (HH * WW)            /* 4096  */
#define EPSV 0.1f

// center/err per element:
//   a = relu(EPS - x)*0.5 ; b = relu(x - (1-EPS))*0.5
//   center = x + a - b    ; err = EPS - a - b
__device__ __forceinline__ void center_err(float x, float& cen, float& er) {
    float a = fmaxf(EPSV - x, 0.0f) * 0.5f;
    float b = fmaxf(x - (1.0f - EPSV), 0.0f) * 0.5f;
    cen = x + a - b;
    er  = EPSV - a - b;
}

// ---------------------------------------------------------------------------
// Kernel 1: single workgroup (1024 threads = 32 wave32 waves).
// Writes zono row 0 (center), terms (as floats), col_of_row[], count.
// Block-wide exclusive scan of mask counts: per-wave __shfl_up inclusive scan
// + LDS scan of 32 wave sums (wave32-aware; warpSize == 32 on gfx1250).
// ---------------------------------------------------------------------------
__global__ void __launch_bounds__(1024)
zono_scan_kernel(const float* __restrict__ x,
                 float* __restrict__ out,         // zono | terms
                 int* __restrict__ col_of_row,    // [N+1]
                 int* __restrict__ count)         // [1]
{
    const int tid   = threadIdx.x;        // 0..1023
    const int CH    = NN / 1024;          // 12 contiguous elements per thread
    const int base  = tid * CH;

    // Pass 1: center -> zono row 0; stash err; count mask hits in my chunk.
    float er[CH];
    int cnt = 0;
    #pragma unroll
    for (int k = 0; k < CH; ++k) {
        float cen, e;
        center_err(x[base + k], cen, e);
        out[base + k] = cen;              // zono row 0
        er[k] = e;
        cnt += (e >= 0.0f) ? 1 : 0;
    }

    // Wave-level inclusive scan (wave32).
    const int lane = tid & 31;
    const int wave = tid >> 5;
    int inc = cnt;
    #pragma unroll
    for (int d = 1; d < 32; d <<= 1) {
        int n = __shfl_up(inc, d, 32);
        if (lane >= d) inc += n;
    }

    __shared__ int wsum[32];
    __shared__ int wbase[32];
    if (lane == 31) wsum[wave] = inc;
    __syncthreads();

    if (wave == 0) {
        int v  = wsum[lane];
        int vi = v;
        #pragma unroll
        for (int d = 1; d < 32; d <<= 1) {
            int n = __shfl_up(vi, d, 32);
            if (lane >= d) vi += n;
        }
        wbase[lane] = vi - v;             // exclusive wave prefix
        if (lane == 31) *count = vi;      // total masked elements
    }
    __syncthreads();

    // Exclusive prefix for this thread's chunk.
    int running = wbase[wave] + (inc - cnt);

    // Pass 2: assign 1-based rows, record inverse map, write terms as floats.
    float* terms = out + (size_t)(NN + 1) * (size_t)NN;
    #pragma unroll
    for (int k = 0; k < CH; ++k) {
        const int i = base + k;
        if (er[k] >= 0.0f) {
            ++running;                    // inclusive cumsum value at i
            col_of_row[running] = i;
            terms[2 * i + 0] = (float)running;
            terms[2 * i + 1] = (float)(i / HWW);
        } else {
            terms[2 * i + 0] = -1.0f;
            terms[2 * i + 1] = -1.0f;
        }
    }
}

// ---------------------------------------------------------------------------
// Kernel 2: one block per zono row r = 1..N (row 0 done by kernel 1).
// Zero-fill the 48KB row with ASYNC LDS->global B128 stores (CDNA5 path,
// ASYNCcnt-tracked, non-temporal so 604MB doesn't thrash the 192MB L2),
// then scatter this row's single err value (if any) after s_wait_asynccnt.
// ---------------------------------------------------------------------------
__global__ void __launch_bounds__(256)
zono_fill_kernel(const float* __restrict__ x,
                 float* __restrict__ out,
                 const int* __restrict__ col_of_row,
                 const int* __restrict__ count)
{
    const int r   = blockIdx.x + 1;       // rows 1..N
    const int tid = threadIdx.x;          // 0..255
    float* row = out + (size_t)r * (size_t)NN;

    // 4KB of zeros in LDS, 16B per lane.
    __shared__ __align__(16) float zsrc[256 * 4];
    float4 z; z.x = 0.f; z.y = 0.f; z.z = 0.f; z.w = 0.f;
    *(float4*)&zsrc[tid * 4] = z;
    __syncthreads();                      // ds writes visible before async engine reads LDS

    // Low 32 bits of a flat pointer into the LDS aperture == LDS byte offset.
    const unsigned lds_off = (unsigned)(uintptr_t)&zsrc[tid * 4];
    const unsigned long long gbase = (unsigned long long)(uintptr_t)row;

    // 12 iterations x 256 lanes x 16B = 49152B = one full row of zeros.
    #pragma unroll
    for (int it = 0; it < 12; ++it) {
        unsigned long long ga =
            gbase + ((unsigned long long)(it * 256 + tid)) * 16ull;
        asm volatile("global_store_async_from_lds_b128 %0, %1, off th:TH_STORE_NT"
                     :: "v"(ga), "v"(lds_off)
                     : "memory");
    }
    // Drain this wave's async stores before the scatter write can land.
    asm volatile("s_wait_asynccnt 0" ::: "memory");
    __syncthreads();                      // all waves' zeros are in memory now

    if (tid == 0 && r <= *count) {
        const int c = col_of_row[r];
        float cen, e;
        center_err(x[c], cen, e);         // recompute err (x is never mutated)
        row[c] = e;
    }
}

// ---------------------------------------------------------------------------
extern "C" void kernel_launch(void* const* d_in, const int* in_sizes, int n_in,
                              void* d_out, int out_size, void* d_ws, size_t ws_size,
                              hipStream_t stream)
{
    (void)in_sizes; (void)n_in; (void)out_size; (void)ws_size;
    const float* x = (const float*)d_in[0];
    float* out = (float*)d_out;

    // Workspace: [0] count (int), [16..] col_of_row[N+1] (ints), ~49KB total.
    int* count      = (int*)d_ws;
    int* col_of_row = (int*)((char*)d_ws + 16);

    zono_scan_kernel<<<1, 1024, 0, stream>>>(x, out, col_of_row, count);
    zono_fill_kernel<<<NN, 256, 0, stream>>>(x, out, col_of_row, count);
}